// NewtonNet_39539468927639
// MI455X (gfx1250) — compile-verified
//
#include <hip/hip_runtime.h>
#include <math.h>

// ---------------------------------------------------------------------------
// NewtonNet forward for MI455X (gfx1250, wave32, WMMA f32 16x16x4,
// async global->LDS staging)
// ---------------------------------------------------------------------------

#define NN_NODES 10000
#define NN_EDGES 160000
#define FD 128
#define NB 20
#define NL 3
#define CUT 5.0f

typedef float v2f __attribute__((ext_vector_type(2)));
typedef float v8f __attribute__((ext_vector_type(8)));

// ---------------------------------------------------------------------------
// GEMM: C[M x 128] = act(A[M x 128] @ B[128 x 128] + bias)
// One block = 256 threads = 8 waves; block covers 128 rows of A.
// Each wave computes a 16-row strip across all 128 columns:
//   8 C-tiles of 16x16, accumulated with v_wmma_f32_16x16x4_f32.
// B (64KB) is staged into LDS in WMMA-fragment order via
// GLOBAL_LOAD_ASYNC_TO_LDS_B32 (per-lane LDS dest => swizzle for free),
// so the inner loop does one conflict-free ds_load_b64 per fragment.
// act: 0 = identity, 1 = SiLU
// ---------------------------------------------------------------------------
__global__ __launch_bounds__(256) void gemm128_wmma(
    const float* __restrict__ A, const float* __restrict__ B,
    float* __restrict__ C, const float* __restrict__ bias,
    int M, int act)
{
    __shared__ float Bs[32 * 8 * 32 * 2];   // [ks][tile][lane][2] = 64 KB

    const int tid  = threadIdx.x;
    const int lane = tid & 31;
    const int wave = tid >> 5;

    // ---- stage B swizzled: pair p = (ks*8 + t)*32 + lane_s holds
    //      B[4*ks + koff + 0][col], B[4*ks + koff + 1][col]
    //      koff = (lane_s>>4)*2, col = 16*t + (lane_s & 15)
    // Async copy straight into LDS; flat-aperture addr[31:0] == LDS offset.
    for (int p = tid; p < 8192; p += 256) {
        int lane_s = p & 31;
        int t      = (p >> 5) & 7;
        int ks     = p >> 8;
        int koff   = (lane_s >> 4) << 1;
        int col    = (t << 4) + (lane_s & 15);
        int krow   = (ks << 2) + koff;
        unsigned long long g0 = (unsigned long long)(B + krow * FD + col);
        unsigned long long g1 = g0 + (unsigned long long)(FD * sizeof(float));
        unsigned l0 = (unsigned)(uintptr_t)&Bs[p * 2];
        unsigned l1 = l0 + 4u;
        asm volatile("global_load_async_to_lds_b32 %0, %1, off"
                     :: "v"(l0), "v"(g0) : "memory");
        asm volatile("global_load_async_to_lds_b32 %0, %1, off"
                     :: "v"(l1), "v"(g1) : "memory");
    }
    asm volatile("s_wait_asynccnt 0x0" ::: "memory");
    __syncthreads();

    const int m0   = blockIdx.x * 128 + wave * 16;
    const int arow = m0 + (lane & 15);
    const int ak   = (lane >> 4) << 1;               // 0 or 2
    // Clamp OOB rows to a valid row: their accumulators are never stored,
    // so garbage there is harmless and we avoid EXEC-divergent loads.
    const int crow = (arow < M) ? arow : (M - 1);
    const float* Aptr = A + (size_t)crow * FD + ak;

    v8f acc[8];
#pragma unroll
    for (int t = 0; t < 8; ++t)
        acc[t] = (v8f){0.f, 0.f, 0.f, 0.f, 0.f, 0.f, 0.f, 0.f};

    for (int ks = 0; ks < 32; ++ks) {
        v2f afrag = *(const v2f*)(Aptr + (ks << 2));  // global_load_b64
#pragma unroll
        for (int t = 0; t < 8; ++t) {
            v2f bfrag = *(const v2f*)(&Bs[(((ks << 3) + t) * 32 + lane) * 2]); // ds_load_b64
            acc[t] = __builtin_amdgcn_wmma_f32_16x16x4_f32(
                false, afrag, false, bfrag, (short)0, acc[t], false, false);
        }
    }

    // ---- epilogue: bias + activation + store (C layout: VGPR v -> row v / v+8)
    const int col_lo = lane & 15;
    const int rbase  = m0 + ((lane >> 4) << 3);
#pragma unroll
    for (int t = 0; t < 8; ++t) {
        int   col = (t << 4) + col_lo;
        float bv  = bias ? bias[col] : 0.f;
#pragma unroll
        for (int v = 0; v < 8; ++v) {
            int   row = rbase + v;
            float x   = acc[t][v] + bv;
            if (act) x = x / (1.f + __expf(-x));      // SiLU
            if (row < M) C[(size_t)row * FD + col] = x;
        }
    }
}

// ---------------------------------------------------------------------------
// atom = node_emb[z]; force = 0
// ---------------------------------------------------------------------------
__global__ void init_nodes(const int* __restrict__ z, const float* __restrict__ emb,
                           float* __restrict__ atom, float* __restrict__ force)
{
    int idx = blockIdx.x * blockDim.x + threadIdx.x;
    if (idx < NN_NODES * FD)
        atom[idx] = emb[z[idx >> 7] * FD + (idx & 127)];
    if (idx < NN_NODES * 3 * FD)
        force[idx] = 0.f;
}

// ---------------------------------------------------------------------------
// Gaussian RBF x cosine cutoff + unit direction vectors
// (cell/batch: disp is identity in plain forward, so pos_d == pos)
// ---------------------------------------------------------------------------
__global__ void edge_embed(const float* __restrict__ pos, const int* __restrict__ ei,
                           float* __restrict__ dist, float* __restrict__ dir)
{
    int e = blockIdx.x * blockDim.x + threadIdx.x;
    if (e >= NN_EDGES) return;
    int i = ei[e], j = ei[NN_EDGES + e];
    float dx = pos[j * 3 + 0] - pos[i * 3 + 0];
    float dy = pos[j * 3 + 1] - pos[i * 3 + 1];
    float dz = pos[j * 3 + 2] - pos[i * 3 + 2];
    float d  = sqrtf(dx * dx + dy * dy + dz * dz + 1e-12f);
    float inv = 1.f / d;
    dir[e * 3 + 0] = dx * inv;
    dir[e * 3 + 1] = dy * inv;
    dir[e * 3 + 2] = dz * inv;
    float fc = (d < CUT) ? 0.5f * (__cosf(3.14159265358979f * d / CUT) + 1.f) : 0.f;
    const float width = CUT / (float)NB;         // 0.25
    const float step  = CUT / (float)(NB - 1);   // linspace spacing
#pragma unroll
    for (int b = 0; b < NB; ++b) {
        float u = (d - step * (float)b) / width;
        dist[e * NB + b] = __expf(-u * u) * fc;
    }
}

// ---------------------------------------------------------------------------
// msg = (dist_edge @ me_W) * h[i] * h[j];  atom += segment_sum(msg, i)
// ---------------------------------------------------------------------------
__global__ void edge_msg(const float* __restrict__ dist, const int* __restrict__ ei,
                         const float* __restrict__ h, const float* __restrict__ meW,
                         float* __restrict__ msg, float* __restrict__ atom)
{
    int idx = blockIdx.x * blockDim.x + threadIdx.x;
    if (idx >= NN_EDGES * FD) return;
    int e = idx >> 7, f = idx & 127;
    int i = ei[e], j = ei[NN_EDGES + e];
    const float* de = dist + e * NB;
    float me = 0.f;
#pragma unroll
    for (int b = 0; b < NB; ++b) me += de[b] * meW[b * FD + f];
    float m = me * h[i * FD + f] * h[j * FD + f];
    msg[idx] = m;
    atomicAdd(&atom[i * FD + f], m);
}

// ---------------------------------------------------------------------------
// dforce[i] += t1*dir[e] + t2*force_old[j]   (scatter into fresh accumulator)
// ---------------------------------------------------------------------------
__global__ void edge_force(const int* __restrict__ ei, const float* __restrict__ t1,
                           const float* __restrict__ t2, const float* __restrict__ dir,
                           const float* __restrict__ force, float* __restrict__ dforce)
{
    int idx = blockIdx.x * blockDim.x + threadIdx.x;
    if (idx >= NN_EDGES * FD) return;
    int e = idx >> 7, f = idx & 127;
    int i = ei[e], j = ei[NN_EDGES + e];
    float v1 = t1[idx], v2 = t2[idx];
#pragma unroll
    for (int d = 0; d < 3; ++d) {
        float val = v1 * dir[e * 3 + d] + v2 * force[(size_t)(j * 3 + d) * FD + f];
        atomicAdd(&dforce[(size_t)(i * 3 + d) * FD + f], val);
    }
}

__global__ void zero_buf(float* __restrict__ p, int n)
{
    int i = blockIdx.x * blockDim.x + threadIdx.x;
    if (i < n) p[i] = 0.f;
}

__global__ void add_buf(float* __restrict__ dst, const float* __restrict__ src, int n)
{
    int i = blockIdx.x * blockDim.x + threadIdx.x;
    if (i < n) dst[i] += src[i];
}

// atom[n] += sum_d force[n,d,:] * g[n,d,:]
__global__ void atom_update(const float* __restrict__ force, const float* __restrict__ g,
                            float* __restrict__ atom)
{
    int idx = blockIdx.x * blockDim.x + threadIdx.x;
    if (idx >= NN_NODES * FD) return;
    int n = idx >> 7, f = idx & 127;
    float s = 0.f;
#pragma unroll
    for (int d = 0; d < 3; ++d)
        s += force[(size_t)(n * 3 + d) * FD + f] * g[(size_t)(n * 3 + d) * FD + f];
    atom[idx] += s;
}

// ---------------------------------------------------------------------------
extern "C" void kernel_launch(void* const* d_in, const int* in_sizes, int n_in,
                              void* d_out, int out_size, void* d_ws, size_t ws_size,
                              hipStream_t stream)
{
    const int*   z       = (const int*)  d_in[0];
    const float* pos     = (const float*)d_in[1];
    // d_in[2] cell, d_in[3] batch: identity no-ops in this forward
    const int*   ei      = (const int*)  d_in[4];
    const float* emb     = (const float*)d_in[5];
    const float* mnp_W1  = (const float*)d_in[6];
    const float* mnp_b1  = (const float*)d_in[7];
    const float* mnp_W2  = (const float*)d_in[8];
    const float* mnp_b2  = (const float*)d_in[9];
    const float* me_W    = (const float*)d_in[10];
    const float* em1_W1  = (const float*)d_in[11];
    const float* em1_W2  = (const float*)d_in[12];
    const float* em2_W1  = (const float*)d_in[13];
    const float* em2_W2  = (const float*)d_in[14];
    const float* eu_W    = (const float*)d_in[15];

    float* atom  = (float*)d_out;                    // (N, F)
    float* force = atom + (size_t)NN_NODES * FD;     // (N, 3, F)

    // workspace layout (floats)
    float* ws     = (float*)d_ws;
    size_t o      = 0;
    float* tmp    = ws + o; o += (size_t)NN_NODES * FD;
    float* h      = ws + o; o += (size_t)NN_NODES * FD;
    float* dist   = ws + o; o += (size_t)NN_EDGES * NB;
    float* dir    = ws + o; o += (size_t)NN_EDGES * 3;
    float* msg    = ws + o; o += (size_t)NN_EDGES * FD;
    float* abuf   = ws + o; o += (size_t)NN_EDGES * FD;
    float* t1     = ws + o; o += (size_t)NN_EDGES * FD;
    float* t2     = ws + o; o += (size_t)NN_EDGES * FD;
    float* dforce = ws + o; o += (size_t)NN_NODES * 3 * FD;
    float* g      = ws + o; o += (size_t)NN_NODES * 3 * FD;

    const int TB = 256;
    const int nF   = NN_NODES * FD;            // 1.28M
    const int n3F  = NN_NODES * 3 * FD;        // 3.84M
    const int eF   = NN_EDGES * FD;            // 20.48M
    const int gN   = (NN_NODES + 127) / 128;   // gemm blocks, M = N
    const int gE   = (NN_EDGES + 127) / 128;   // gemm blocks, M = E
    const int g3N  = (3 * NN_NODES + 127) / 128;

    init_nodes<<<(n3F + TB - 1) / TB, TB, 0, stream>>>(z, emb, atom, force);
    edge_embed<<<(NN_EDGES + TB - 1) / TB, TB, 0, stream>>>(pos, ei, dist, dir);

    for (int l = 0; l < NL; ++l) {
        // h = silu(atom @ mnp_W1 + b1) @ mnp_W2 + b2
        gemm128_wmma<<<gN, TB, 0, stream>>>(atom, mnp_W1 + (size_t)l * FD * FD, tmp,
                                            mnp_b1 + (size_t)l * FD, NN_NODES, 1);
        gemm128_wmma<<<gN, TB, 0, stream>>>(tmp, mnp_W2 + (size_t)l * FD * FD, h,
                                            mnp_b2 + (size_t)l * FD, NN_NODES, 0);
        // msg + atom segment-sum
        edge_msg<<<(eF + TB - 1) / TB, TB, 0, stream>>>(dist, ei, h,
                                                        me_W + (size_t)l * NB * FD, msg, atom);
        // t1 = silu(msg @ em1_W1) @ em1_W2
        gemm128_wmma<<<gE, TB, 0, stream>>>(msg, em1_W1 + (size_t)l * FD * FD, abuf,
                                            nullptr, NN_EDGES, 1);
        gemm128_wmma<<<gE, TB, 0, stream>>>(abuf, em1_W2 + (size_t)l * FD * FD, t1,
                                            nullptr, NN_EDGES, 0);
        // t2 = silu(msg @ em2_W1) @ em2_W2
        gemm128_wmma<<<gE, TB, 0, stream>>>(msg, em2_W1 + (size_t)l * FD * FD, abuf,
                                            nullptr, NN_EDGES, 1);
        gemm128_wmma<<<gE, TB, 0, stream>>>(abuf, em2_W2 + (size_t)l * FD * FD, t2,
                                            nullptr, NN_EDGES, 0);
        // force += segment_sum(t1*dir + t2*force_old[j], i)
        zero_buf<<<(n3F + TB - 1) / TB, TB, 0, stream>>>(dforce, n3F);
        edge_force<<<(eF + TB - 1) / TB, TB, 0, stream>>>(ei, t1, t2, dir, force, dforce);
        add_buf<<<(n3F + TB - 1) / TB, TB, 0, stream>>>(force, dforce, n3F);
        // atom += sum_d force * (force @ eu_W)
        gemm128_wmma<<<g3N, TB, 0, stream>>>(force, eu_W + (size_t)l * FD * FD, g,
                                             nullptr, 3 * NN_NODES, 0);
        atom_update<<<(nF + TB - 1) / TB, TB, 0, stream>>>(force, g, atom);
    }
    (void)in_sizes; (void)n_in; (void)out_size; (void)ws_size;
}